// PermutohedralSubmanifoldConvolution_4063039062169
// MI455X (gfx1250) — compile-verified
//
#include <hip/hip_runtime.h>

typedef __attribute__((ext_vector_type(2))) float v2f;
typedef __attribute__((ext_vector_type(8))) float v8f;

#define KF 13   // permutohedral filter volume (DIM*DIM + DIM + 1, DIM=3)

__global__ __launch_bounds__(256) void permuto_conv_wmma(
    const float* __restrict__ features,   // [N, 16]
    const float* __restrict__ weight,     // [KF, 1, 16, 16] row-major
    const float* __restrict__ bias,       // [16]
    const int*   __restrict__ nbr,        // [KF, N]
    float* __restrict__ out,              // [N, 16]
    int N, int numTiles)
{
    const int lane = threadIdx.x & 31;
    const int m    = lane & 15;   // A: tile row / idx row; B,C: output column n
    const int hi   = lane >> 4;   // half-wave selector

    // ---- Preload all weights into registers (loop-invariant B operands).
    // B chunk j of offset k covers K = 4j..4j+3:
    //   lane(n,hi):  .x = W[k][4j+2hi][n], .y = W[k][4j+2hi+1][n]
    // 13 KB total -> L2/L0 resident; default (RT) caching.
    v2f B[KF][4];
#pragma unroll
    for (int k = 0; k < KF; ++k) {
#pragma unroll
        for (int j = 0; j < 4; ++j) {
            const int ci = 4 * j + 2 * hi;
            const float* wp = weight + k * 256 + ci * 16 + m;
            v2f b;
            b.x = wp[0];
            b.y = wp[16];   // next ci row, same column
            B[k][j] = b;
        }
    }

    const float bv = bias[m];

    const int wavesPerBlock = blockDim.x >> 5;
    const int waveId   = blockIdx.x * wavesPerBlock + (threadIdx.x >> 5);
    const int numWaves = gridDim.x * wavesPerBlock;

    for (int t = waveId; t < numTiles; t += numWaves) {
        const int base = t * 16;
        v8f acc = {bv, bv, bv, bv, bv, bv, bv, bv};

#pragma unroll
        for (int k = 0; k < KF; ++k) {
            // Streaming read-once index: non-temporal so it does not evict
            // the 128MB features working set from the 192MB L2.
            const int row = __builtin_nontemporal_load(nbr + k * N + base + m);
            const float* src = features + (long)row * 16 + 2 * hi;

            // A chunks (regular temporal: keep features L2-resident):
            // lane(m,hi) holds A[m, 4j+2hi] and A[m, 4j+2hi+1]
            v2f a0 = *(const v2f*)(src + 0);
            v2f a1 = *(const v2f*)(src + 4);
            v2f a2 = *(const v2f*)(src + 8);
            v2f a3 = *(const v2f*)(src + 12);

            acc = __builtin_amdgcn_wmma_f32_16x16x4_f32(false, a0, false, B[k][0], (short)0, acc, false, false);
            acc = __builtin_amdgcn_wmma_f32_16x16x4_f32(false, a1, false, B[k][1], (short)0, acc, false, false);
            acc = __builtin_amdgcn_wmma_f32_16x16x4_f32(false, a2, false, B[k][2], (short)0, acc, false, false);
            acc = __builtin_amdgcn_wmma_f32_16x16x4_f32(false, a3, false, B[k][3], (short)0, acc, false, false);
        }

        // C/D layout: lane(n,hi), VGPR r -> out[base + r + 8*hi][n]
        // Write-once output: non-temporal stores, don't pollute L2.
#pragma unroll
        for (int r = 0; r < 8; ++r) {
            __builtin_nontemporal_store(acc[r],
                out + (long)(base + r + 8 * hi) * 16 + m);
        }
    }

    // ---- Scalar tail (N % 16 != 0). Not taken for N = 2,000,000. No WMMA here.
    const int tailStart = numTiles * 16;
    if (tailStart < N) {
        const int gtid  = blockIdx.x * blockDim.x + threadIdx.x;
        const int total = gridDim.x * blockDim.x;
        const int tailElems = (N - tailStart) * 16;
        for (int e = gtid; e < tailElems; e += total) {
            const int site = tailStart + (e >> 4);
            const int ch   = e & 15;
            float s = bias[ch];
            for (int k = 0; k < KF; ++k) {
                const int row = nbr[k * N + site];
                const float* fr = features + (long)row * 16;
                const float* wr = weight + k * 256 + ch;
                for (int ci = 0; ci < 16; ++ci)
                    s += fr[ci] * wr[ci * 16];
            }
            out[(long)site * 16 + ch] = s;
        }
    }
}

extern "C" void kernel_launch(void* const* d_in, const int* in_sizes, int n_in,
                              void* d_out, int out_size, void* d_ws, size_t ws_size,
                              hipStream_t stream) {
    const float* features = (const float*)d_in[0];   // [N,16] f32
    const float* weight   = (const float*)d_in[1];   // [13,1,16,16] f32
    const float* bias     = (const float*)d_in[2];   // [16] f32
    const int*   nbr      = (const int*)d_in[3];     // [13,N] i32
    float*       out      = (float*)d_out;           // [N,16] f32

    const int N        = in_sizes[0] / 16;
    const int numTiles = N / 16;

    // 8 waves/block; up to ~16K waves grid-striding over tiles to hide
    // random-gather latency on a very wide part.
    int blocks = (numTiles + 7) / 8;
    if (blocks > 2048) blocks = 2048;
    if (blocks < 1)    blocks = 1;

    permuto_conv_wmma<<<blocks, 256, 0, stream>>>(features, weight, bias, nbr,
                                                  out, N, numTiles);
}